// AddContextFrames_53283364274940
// MI455X (gfx1250) — compile-verified
//
#include <hip/hip_runtime.h>
#include <hip/hip_bf16.h>
#include <stdint.h>

// AddContextFrames: out[0, w*80+f, t] = signal[0, f, t + w - 9], zero-padded.
// Pure data movement: ~10.5 MB read, ~199 MB written -> HBM-store-bound, ~9us floor at 23.3 TB/s.
// CDNA5 path: TDM (tensor_load_to_lds) stages the input tile into LDS asynchronously,
// then 19 shifted rows are fanned out with coalesced non-temporal b128 stores.
//
// Toolchain note: this lane is amdgpu-toolchain (clang-23) -> 6-arg tensor_load_to_lds:
// (uint32x4 g0, int32x8 g1, int32x4 g2, int32x4 g3, int32x8 g4, i32 cpol).

#define STEPS 32768
#define FEATS 80
#define NCTX  9
#define WIN   19
#define TILE  2048   // columns per workgroup; LDS = (2048+18)*4 = 8264 B

#ifndef USE_TDM
#define USE_TDM 1
#endif

typedef __attribute__((ext_vector_type(4))) unsigned int u32x4;
typedef __attribute__((ext_vector_type(8))) int          i32x8;
typedef __attribute__((ext_vector_type(4))) int          i32x4;
typedef __attribute__((ext_vector_type(4))) float        f32x4;

__global__ __launch_bounds__(256)
void add_ctx_frames_kernel(const float* __restrict__ in, float* __restrict__ out) {
    __shared__ __align__(16) float lds[TILE + 2 * NCTX];

    const int tid = threadIdx.x;
    const int t0  = blockIdx.x * TILE;   // start column of this tile
    const int f   = blockIdx.y;          // feature row
    const float* __restrict__ row = in + (size_t)f * STEPS;

    // ---- Halo: 9 front + 9 back elements, zero-filled out of range ----
    // lds[i] corresponds to in[f, t0 - 9 + i]
    if (tid < 2 * NCTX) {
        const int i = (tid < NCTX) ? tid : (TILE + tid);  // 0..8 front, 2057..2065 back
        const int t = t0 - NCTX + i;
        lds[i] = (t >= 0 && t < STEPS) ? row[t] : 0.0f;
    }

    // ---- Interior: TDM async DMA of 2048 floats -> lds[9 .. 2057) ----
#if USE_TDM && __has_builtin(__builtin_amdgcn_tensor_load_to_lds) && __has_builtin(__builtin_amdgcn_s_wait_tensorcnt)
    if (tid < 32) {  // single wave issues the tensor op (TDM is per-wave, EXEC-independent)
        const uint64_t ga = (uint64_t)(uintptr_t)(row + t0);       // tile start, fully in-bounds
        const uint32_t la = (uint32_t)(uintptr_t)(&lds[NCTX]);     // LDS byte address (low 32 bits of flat addr)

        u32x4 g0;                      // D# group 0 (ISA 8.3)
        g0[0] = 1u;                                                 // count=1, is_restore=0, gather=0
        g0[1] = la;                                                 // lds_addr [63:32]
        g0[2] = (uint32_t)(ga & 0xFFFFFFFFu);                       // global_addr [95:64]
        g0[3] = (uint32_t)((ga >> 32) & 0x01FFFFFFu) | (2u << 30);  // global_addr hi + type=2 (image)

        i32x8 g1;                      // D# group 1 (ISA 8.4)
        g1[0] = (int)(2u << 16);                         // wg_mask=0, data_size=2 (4 bytes)
        g1[1] = (int)((uint32_t)(TILE & 0xFFFF) << 16);  // atomic_barrier_addr=0, tensor_dim0[15:0]=2048
        g1[2] = (int)(1u << 16);                         // tensor_dim0[31:16]=0, tensor_dim1=1
        g1[3] = (int)((uint32_t)TILE << 16);             // tensor_dim1 hi=0, tile_dim0=2048
        g1[4] = 1;                                       // tile_dim1=1, tile_dim2=0
        g1[5] = TILE;                                    // tensor_dim0_stride = 2048
        g1[6] = 0;                                       // stride0 hi, tensor_dim1_stride lo
        g1[7] = 0;                                       // tensor_dim1_stride hi

        i32x4 g2 = {0, 0, 0, 0};       // <=2D tensor: groups 2/3 unused
        i32x4 g3 = {0, 0, 0, 0};
        i32x8 g4 = {0, 0, 0, 0, 0, 0, 0, 0};  // extra group required by 6-arg builtin; unused here

        __builtin_amdgcn_tensor_load_to_lds(g0, g1, g2, g3, g4, 0);
        __builtin_amdgcn_s_wait_tensorcnt(0);            // drain TENSORcnt before barrier signal
    }
#else
    for (int i = tid; i < TILE; i += 256) {
        lds[NCTX + i] = row[t0 + i];
    }
#endif

    __syncthreads();

    // ---- Fan-out: 19 shifted rows, 2 float4 per thread per row (fully coalesced) ----
    // out[(w*80+f), t0+tloc + k] = in[f, t0+tloc + k + w - 9] = lds[tloc + w + k]
#pragma unroll
    for (int c = 0; c < 2; ++c) {
        const int tloc = c * 1024 + tid * 4;   // 16B-aligned LDS byte offset -> ds_load_b128-able

        float r[WIN + 3];                      // sliding 22-float register window
#pragma unroll
        for (int k = 0; k < WIN + 3; ++k) {
            r[k] = lds[tloc + k];
        }

#pragma unroll
        for (int w = 0; w < WIN; ++w) {
            f32x4 v;
            v.x = r[w + 0];
            v.y = r[w + 1];
            v.z = r[w + 2];
            v.w = r[w + 3];
            float* __restrict__ orow = out + ((size_t)(w * FEATS + f) << 15) + t0 + tloc;
            __builtin_nontemporal_store(v, (f32x4*)orow);   // write-once output: TH=NT
        }
    }
}

extern "C" void kernel_launch(void* const* d_in, const int* in_sizes, int n_in,
                              void* d_out, int out_size, void* d_ws, size_t ws_size,
                              hipStream_t stream) {
    (void)in_sizes; (void)n_in; (void)d_ws; (void)ws_size; (void)out_size;
    const float* in = (const float*)d_in[0];
    float* out      = (float*)d_out;

    dim3 grid(STEPS / TILE, FEATS, 1);   // 16 x 80 = 1280 workgroups
    dim3 block(256, 1, 1);               // 8 wave32s per WGP-resident block
    add_ctx_frames_kernel<<<grid, block, 0, stream>>>(in, out);
}